// LocalDiffusionInteractionBlock_2370821947766
// MI455X (gfx1250) — compile-verified
//
#include <hip/hip_runtime.h>
#include <math.h>

// ---------------------------------------------------------------------------
// MI455X (gfx1250) implementation — fp32 WMMA (V_WMMA_F32_16X16X4_F32).
// Problem is HBM-bound (~0.7GB @ 23.3TB/s); fp32 matrix cores exceed the
// roofline, so no precision downcast. GEMM is templated on K so the K-chunk
// loop fully unrolls: 2 WMMAs per k-step, all operands from LDS.
// Workspace (floats), ~651 MB:
//   node_scalars N*128 | s_up N*128 | v_up 3*N*128 | W1pad 272*64 |
//   wiN E*272 | h1 E*64 | h2 E*64 | tpw E*512 | M0 N*256 | M1 3*N*256
// ---------------------------------------------------------------------------

typedef float v2f __attribute__((ext_vector_type(2)));
typedef float v8f __attribute__((ext_vector_type(8)));

// C = silu?( scale * A(MxK) @ B(KxN) ), strided A-K (for the v_in einsum) and
// strided C (for the interleaved (N,128,4) output store).
// Block: 256 threads = 8 waves. Block tile 64(M) x 64(N).
// Wave tile 16(M) x 32(N): two 16x16 accumulators sharing one A fragment.
// K and KC are compile-time; K % KC == 0 for every instantiation used.
template<int K, int KC>
__global__ __launch_bounds__(256)
void gemm_wmma_f32(const float* __restrict__ A, int lda, int astride,
                   const float* __restrict__ B, int ldb,
                   float* __restrict__ C, int ldcRow, int cstride,
                   int M, float scale, int do_silu)
{
    static_assert(K % KC == 0 && KC % 4 == 0, "bad K chunking");
    constexpr int LDA = KC + 2;   // even stride -> 8B-aligned paired A reads
    constexpr int LDB = 68;       // even stride for the 64-wide B panel
    __shared__ float sA[64][LDA];
    __shared__ float sB[KC][LDB];

    const int tid  = threadIdx.x;
    const int wave = tid >> 5;
    const int lane = tid & 31;
    const int hi   = lane >> 4;       // half-wave: K+2 (A/B), M+8 (C/D)
    const int nl   = lane & 15;

    const int m0 = (wave >> 1) * 16;          // 0,16,32,48
    const int n0 = (wave & 1) * 32;           // 0,32
    const int rowBase = blockIdx.x * 64;
    const int colBase = blockIdx.y * 64;

    v8f acc0 = {}, acc1 = {};

    for (int k0 = 0; k0 < K; k0 += KC) {
        __syncthreads();
        // Stage A panel (64 x KC) into LDS.
        #pragma unroll
        for (int idx = tid; idx < 64 * KC; idx += 256) {
            const int r  = idx / KC;
            const int kk = idx - r * KC;
            const int grow = rowBase + r;
            sA[r][kk] = (grow < M)
                ? A[(size_t)grow * lda + (size_t)(k0 + kk) * astride]
                : 0.0f;
        }
        // Stage B panel (KC x 64) into LDS — coalesced 64-wide rows.
        #pragma unroll
        for (int idx = tid; idx < KC * 64; idx += 256) {
            const int r  = idx >> 6;
            const int cc = idx & 63;
            sB[r][cc] = B[(size_t)(k0 + r) * ldb + colBase + cc];
        }
        __syncthreads();

        // Fully unrolled: KC/4 k-steps, 2 WMMAs each, all operands from LDS.
        #pragma unroll
        for (int kk = 0; kk < KC; kk += 4) {
            v2f a, b0, b1;
            // A 16x4 fragment: lane row = lane&15; VGPR pair = K,K+1 with
            // lanes 16-31 shifted +2 in K (ISA layout).
            a.x = sA[m0 + nl][kk + 2 * hi];
            a.y = sA[m0 + nl][kk + 2 * hi + 1];
            // B 4x16 fragments (mirror layout), two N sub-tiles.
            b0.x = sB[kk + 2 * hi][n0 + nl];
            b0.y = sB[kk + 2 * hi + 1][n0 + nl];
            b1.x = sB[kk + 2 * hi][n0 + 16 + nl];
            b1.y = sB[kk + 2 * hi + 1][n0 + 16 + nl];
            acc0 = __builtin_amdgcn_wmma_f32_16x16x4_f32(
                false, a, false, b0, (short)0, acc0, false, false);
            acc1 = __builtin_amdgcn_wmma_f32_16x16x4_f32(
                false, a, false, b1, (short)0, acc1, false, false);
        }
    }

    // C/D layout: VGPR r -> M = r + 8*hi within tile, N = lane&15.
    #pragma unroll
    for (int t = 0; t < 2; ++t) {
        const v8f acc = t ? acc1 : acc0;
        const int gn = colBase + n0 + 16 * t + nl;
        #pragma unroll
        for (int r = 0; r < 8; ++r) {
            const int grow = rowBase + m0 + r + 8 * hi;
            if (grow < M) {
                float x = acc[r] * scale;
                if (do_silu)
                    x = x * __builtin_amdgcn_rcpf(1.0f + __expf(-x));
                C[(size_t)grow * ldcRow + (size_t)gn * cstride] = x;
            }
        }
    }
}

// Zero M0/M1 accumulators and build zero-padded W1 (272x64).
__global__ void init_ws_kernel(float* __restrict__ zero_base, size_t nz,
                               const float* __restrict__ W1,
                               float* __restrict__ W1p)
{
    size_t i = (size_t)blockIdx.x * blockDim.x + threadIdx.x;
    size_t stride = (size_t)gridDim.x * blockDim.x;
    for (size_t j = i; j < nz; j += stride) zero_base[j] = 0.0f;
    for (size_t j = i; j < (size_t)272 * 64; j += stride) {
        int row = (int)(j >> 6);           // /64
        W1p[j] = (row < 265) ? W1[j] : 0.0f;
    }
}

// One wave per edge: build wi row (265 cols), switch-norm, write padded 272.
__global__ __launch_bounds__(256)
void edge_prep_kernel(const float* __restrict__ ns,      // node_scalars N*128
                      const float* __restrict__ elen,    // E
                      const float* __restrict__ tdiff,   // E
                      const float* __restrict__ lens,    // E ("lengths")
                      const int*   __restrict__ eidx,    // 2*E
                      const float* __restrict__ snw, const float* __restrict__ snb,
                      const float* __restrict__ mw,  const float* __restrict__ vwv,
                      const float* __restrict__ rm,  const float* __restrict__ rv,
                      float* __restrict__ wiN, int E)
{
    const int wave = (int)((blockIdx.x * blockDim.x + threadIdx.x) >> 5);
    const int lane = threadIdx.x & 31;
    if (wave >= E) return;
    const int e = wave;

    const float r  = elen[e];
    const float tv = tdiff[e];
    const int snd = eidx[e];
    const int rcv = eidx[E + e];

    // poly cutoff, p=5: f = 1 - 21u^5 + 35u^6 - 15u^7, u<1
    const float u  = r * (1.0f / 5.0f);
    const float u2 = u * u;
    const float u5 = u2 * u2 * u;
    float f = 1.0f - 21.0f * u5 + 35.0f * u5 * u - 15.0f * u5 * u2;
    f = (u < 1.0f) ? f : 0.0f;

    const float PI_R = 3.14159265358979323846f / 5.0f;  // pi/R_MAX
    const float PREF = 0.63245553203367587f;            // sqrt(2/5)
    const float rinv = __builtin_amdgcn_rcpf(r);

    float vals[9];
    float s1 = 0.0f, s2 = 0.0f;
    #pragma unroll
    for (int i = 0; i < 9; ++i) {
        const int c = lane + 32 * i;
        float x = 0.0f;
        if (c < 128) {
            x = ns[(size_t)snd * 128 + c] * f;
        } else if (c < 256) {
            x = ns[(size_t)rcv * 128 + (c - 128)] * f;
        } else if (c < 264) {
            const float kf = (float)(c - 255) * PI_R;
            x = PREF * __sinf(kf * r) * rinv * __expf(-kf * kf * tv) * f;
        } else if (c == 264) {
            x = lens[e] * f;
        }
        vals[i] = x;
        if (c < 265) { s1 += x; s2 += x * x; }
    }
    #pragma unroll
    for (int off = 16; off > 0; off >>= 1) {
        s1 += __shfl_xor(s1, off, 32);
        s2 += __shfl_xor(s2, off, 32);
    }
    const float mean_ln = s1 * (1.0f / 265.0f);
    const float var_ln  = (s2 - 265.0f * mean_ln * mean_ln) * (1.0f / 264.0f);

    // softmax over the 2-element switch weights
    float a0 = mw[0], a1 = mw[1];
    float am = fmaxf(a0, a1);
    float e0 = __expf(a0 - am), e1 = __expf(a1 - am);
    float msum = __builtin_amdgcn_rcpf(e0 + e1);
    const float ms0 = e0 * msum, ms1 = e1 * msum;
    float b0 = vwv[0], b1 = vwv[1];
    float bm = fmaxf(b0, b1);
    float f0 = __expf(b0 - bm), f1 = __expf(b1 - bm);
    float vsum = __builtin_amdgcn_rcpf(f0 + f1);
    const float vs0 = f0 * vsum, vs1 = f1 * vsum;

    float* row = wiN + (size_t)e * 272;
    #pragma unroll
    for (int i = 0; i < 9; ++i) {
        const int c = lane + 32 * i;
        if (c < 265) {
            const float mean = ms0 * mean_ln + ms1 * rm[c];
            const float var  = vs0 * var_ln + vs1 * rv[c];
            row[c] = (vals[i] - mean) * rsqrtf(var + 1e-5f) * snw[c] + snb[c];
        } else if (c < 272) {
            row[c] = 0.0f;   // zero-pad so K=272 GEMM is exact
        }
    }
}

// One wave per edge: build messages and atomically segment-sum into M0/M1.
__global__ __launch_bounds__(256)
void scatter_kernel(const float* __restrict__ tpw,   // E*512
                    const float* __restrict__ sup,   // N*128
                    const float* __restrict__ vup,   // 3 planes of N*128
                    const float* __restrict__ eattr, // E*4
                    const int*   __restrict__ eidx,  // 2*E
                    float* __restrict__ M0,          // N*256
                    float* __restrict__ M1,          // 3 planes of N*256
                    int E, int N)
{
    const int wave = (int)((blockIdx.x * blockDim.x + threadIdx.x) >> 5);
    const int lane = threadIdx.x & 31;
    if (wave >= E) return;
    const int e = wave;

    const int snd = eidx[e];
    const int rcv = eidx[E + e];
    const float y0 = eattr[(size_t)e * 4 + 0];
    const float yx = eattr[(size_t)e * 4 + 1];
    const float yy = eattr[(size_t)e * 4 + 2];
    const float yz = eattr[(size_t)e * 4 + 3];
    const float INV_SQ3 = 0.57735026918962576f;

    const float* tp = tpw + (size_t)e * 512;
    const size_t plane = (size_t)N * 256;
    const size_t sbase = (size_t)snd * 128;
    const size_t rbase = (size_t)rcv * 256;
    const size_t vplane = (size_t)N * 128;

    #pragma unroll
    for (int i = 0; i < 4; ++i) {
        const int uch = lane + 32 * i;
        const float w0 = tp[uch];
        const float w1 = tp[128 + uch];
        const float w2 = tp[256 + uch];
        const float w3 = tp[384 + uch];
        const float se  = sup[sbase + uch];
        const float ve0 = vup[0 * vplane + sbase + uch];
        const float ve1 = vup[1 * vplane + sbase + uch];
        const float ve2 = vup[2 * vplane + sbase + uch];
        const float dot = ve0 * yx + ve1 * yy + ve2 * yz;

        atomicAdd(&M0[rbase + uch],       w0 * se * y0);
        atomicAdd(&M0[rbase + 128 + uch], w3 * dot * INV_SQ3);

        const float w1se = w1 * se;
        atomicAdd(&M1[0 * plane + rbase + uch],       w1se * yx);
        atomicAdd(&M1[1 * plane + rbase + uch],       w1se * yy);
        atomicAdd(&M1[2 * plane + rbase + uch],       w1se * yz);
        atomicAdd(&M1[0 * plane + rbase + 128 + uch], w2 * ve0 * y0);
        atomicAdd(&M1[1 * plane + rbase + 128 + uch], w2 * ve1 * y0);
        atomicAdd(&M1[2 * plane + rbase + 128 + uch], w2 * ve2 * y0);
    }
}

template<int K, int KC>
static inline void launch_gemm(const float* A, int lda, int astride,
                               const float* B, int ldb,
                               float* C, int ldcRow, int cstride,
                               int M, int Ncols, float scale, int silu,
                               hipStream_t stream)
{
    dim3 grid((M + 63) / 64, Ncols / 64);
    gemm_wmma_f32<K, KC><<<grid, 256, 0, stream>>>(
        A, lda, astride, B, ldb, C, ldcRow, cstride, M, scale, silu);
}

extern "C" void kernel_launch(void* const* d_in, const int* in_sizes, int n_in,
                              void* d_out, int out_size, void* d_ws, size_t ws_size,
                              hipStream_t stream)
{
    const float* node_feats   = (const float*)d_in[0];
    const float* edge_attrs   = (const float*)d_in[1];
    const float* edge_lengths = (const float*)d_in[2];
    const float* t_in         = (const float*)d_in[3];
    const float* lengths      = (const float*)d_in[4];
    const int*   edge_index   = (const int*)  d_in[5];
    const float* W_scalar     = (const float*)d_in[6];
    const float* W_up0        = (const float*)d_in[7];
    const float* W_up1        = (const float*)d_in[8];
    const float* sn_weight    = (const float*)d_in[9];
    const float* sn_bias      = (const float*)d_in[10];
    const float* sn_mean_w    = (const float*)d_in[11];
    const float* sn_var_w     = (const float*)d_in[12];
    const float* sn_run_mean  = (const float*)d_in[13];
    const float* sn_run_var   = (const float*)d_in[14];
    const float* W1           = (const float*)d_in[15];
    const float* W2           = (const float*)d_in[16];
    const float* W3           = (const float*)d_in[17];
    const float* W4           = (const float*)d_in[18];
    const float* W_out0       = (const float*)d_in[19];
    const float* W_out1       = (const float*)d_in[20];

    const int N = in_sizes[0] / 512;
    const int E = in_sizes[2];
    float* out = (float*)d_out;

    // --- workspace carve-up (floats) ---
    float* ws = (float*)d_ws;
    size_t o = 0;
    float* ns   = ws + o;  o += (size_t)N * 128;        // node_scalars
    float* sup  = ws + o;  o += (size_t)N * 128;        // s_up
    float* vup  = ws + o;  o += (size_t)3 * N * 128;    // v_up (3 planes)
    float* W1p  = ws + o;  o += (size_t)272 * 64;       // padded W1
    float* wiN  = ws + o;  o += (size_t)E * 272;        // normalized wi (padded)
    float* h1   = ws + o;  o += (size_t)E * 64;
    float* h2   = ws + o;  o += (size_t)E * 64;
    float* tpw  = ws + o;  o += (size_t)E * 512;
    float* M0   = ws + o;  o += (size_t)N * 256;
    float* M1   = ws + o;  o += (size_t)3 * N * 256;    // contiguous with M0

    const float inv128  = 1.0f / sqrtf(128.0f);
    const float inv265  = 1.0f / sqrtf(265.0f);
    const float inv256n = (1.0f / 16.0f) / 16.0f;       // 1/sqrt(256)/AVG_NEIGH

    // 0) init: zero M0+M1 (contiguous), pad W1
    init_ws_kernel<<<2048, 256, 0, stream>>>(M0, (size_t)N * 1024, W1, W1p);

    // 1) node-side GEMMs (K=128)
    launch_gemm<128, 64>(node_feats, 512, 1, W_scalar, 128, ns,  128, 1, N, 128, inv128, 0, stream);
    launch_gemm<128, 64>(node_feats, 512, 1, W_up0,    128, sup, 128, 1, N, 128, inv128, 0, stream);
    for (int d = 0; d < 3; ++d)  // v_in[:,u,d] has K-stride 3 in node_feats
        launch_gemm<128, 64>(node_feats + 128 + d, 512, 3, W_up1, 128,
                             vup + (size_t)d * N * 128, 128, 1, N, 128, inv128, 0, stream);

    // 2) per-edge feature build + switch-norm
    edge_prep_kernel<<<(E + 7) / 8, 256, 0, stream>>>(
        ns, edge_lengths, t_in, lengths, edge_index,
        sn_weight, sn_bias, sn_mean_w, sn_var_w, sn_run_mean, sn_run_var,
        wiN, E);

    // 3) edge MLP (fused scale+SiLU)
    launch_gemm<272, 68>(wiN, 272, 1, W1p, 64,  h1,  64,  1, E,  64, inv265, 1, stream);
    launch_gemm< 64, 64>(h1,  64,  1, W2,  64,  h2,  64,  1, E,  64, 0.125f, 1, stream);
    launch_gemm< 64, 64>(h2,  64,  1, W3,  64,  h1,  64,  1, E,  64, 0.125f, 1, stream);
    launch_gemm< 64, 64>(h1,  64,  1, W4, 512, tpw, 512,  1, E, 512, 0.125f, 0, stream);

    // 4) message build + segment-sum (fp32 atomics into L2-resident M0/M1)
    scatter_kernel<<<(E + 7) / 8, 256, 0, stream>>>(
        tpw, sup, vup, edge_attrs, edge_index, M0, M1, E, N);

    // 5) output GEMMs, strided store into (N,128,4) layout
    launch_gemm<256, 64>(M0, 256, 1, W_out0, 128, out + 0, 512, 4, N, 128, inv256n, 0, stream);
    for (int d = 0; d < 3; ++d)
        launch_gemm<256, 64>(M1 + (size_t)d * N * 256, 256, 1, W_out1, 128,
                             out + 1 + d, 512, 4, N, 128, inv256n, 0, stream);
}